// SortByFirstFeatureAndPool1D_1065151889831
// MI455X (gfx1250) — compile-verified
//
#include <hip/hip_runtime.h>
#include <hip/hip_bf16.h>
#include <stdint.h>

// Problem constants from the reference:
//   inputs: [B=32, N=16384, F=64] float32, top-k with k=2048 on feature 0.
#define BATCHES   32
#define NROWS     16384
#define NFEAT     64
#define KTOP      2048
#define NTHREADS  1024   // 32 wave32 waves, one workgroup per batch / WGP

// One workgroup per batch. Full sort array (16384 x u64 = 128 KB) lives in
// the CDNA5 320KB LDS. Composite key layout (little endian):
//   [63:32] = order-flipped float bits (ascending u64  ==> descending float)
//   [31:0]  = row index                (ascending ties ==> lowest index first)
// which exactly matches jax.lax.top_k ordering.
__global__ __launch_bounds__(NTHREADS)
void topk_sort_gather_kernel(const float* __restrict__ in,
                             float* __restrict__ out) {
    __shared__ uint64_t smem[NROWS];   // 128 KB of the 320 KB WGP LDS

    const unsigned tid = threadIdx.x;
    const unsigned b   = blockIdx.x;
    const float* batch = in + (size_t)b * NROWS * NFEAT;

    // ---- Phase 1: async-copy the strided feature-0 column straight into the
    // HIGH dword of each 8-byte sort slot (GLOBAL_LOAD_ASYNC_TO_LDS_B32,
    // tracked by ASYNCcnt — CDNA5 async data-movement path).
    const unsigned lds_base = (unsigned)(uintptr_t)&smem[0];
    for (unsigned i = tid; i < NROWS; i += NTHREADS) {
        unsigned lds_addr = lds_base + i * 8u + 4u;              // hi dword
        uint64_t gaddr    = (uint64_t)(uintptr_t)(batch + (size_t)i * NFEAT);
        asm volatile("global_load_async_to_lds_b32 %0, %1, off"
                     :: "v"(lds_addr), "v"(gaddr) : "memory");
    }
    // Each thread only re-reads slots it loaded itself, so its own wave's
    // ASYNCcnt drain is sufficient before Phase 2 (barrier comes later).
    asm volatile("s_wait_asynccnt 0" ::: "memory");

    // ---- Phase 2: build composite sort keys in place.
    for (unsigned i = tid; i < NROWS; i += NTHREADS) {
        uint32_t kb = ((uint32_t*)smem)[2u * i + 1u];            // raw float bits
        // monotone float->uint map (ascending):
        uint32_t u  = (kb & 0x80000000u) ? ~kb : (kb | 0x80000000u);
        smem[i] = ((uint64_t)(~u) << 32) | (uint64_t)i;          // ~u => descending
    }
    __syncthreads();

    // ---- Phase 3: LDS-resident bitonic sort, ascending on the composite.
    for (unsigned k = 2; k <= NROWS; k <<= 1) {
        for (unsigned j = k >> 1; j > 0; j >>= 1) {
            for (unsigned i = tid; i < NROWS; i += NTHREADS) {
                unsigned ixj = i ^ j;
                if (ixj > i) {
                    uint64_t a = smem[i];
                    uint64_t c = smem[ixj];
                    bool up = ((i & k) == 0u);
                    if ((a > c) == up) { smem[i] = c; smem[ixj] = a; }
                }
            }
            __syncthreads();
        }
    }

    // ---- Phase 4: gather top-KTOP rows. 16 lanes cooperate per row, one
    // float4 (16B) per lane -> fully coalesced 256B row transfers.
    float* obatch = out + (size_t)b * KTOP * NFEAT;
    const unsigned rows_per_iter = NTHREADS / 16;                // 64
    for (unsigned r0 = 0; r0 < KTOP; r0 += rows_per_iter) {
        unsigned row  = r0 + (tid >> 4);
        unsigned lane = tid & 15u;
        unsigned idx  = (unsigned)(smem[row] & 0xFFFFFFFFu);
        // Prefetch next iteration's data-dependent row (global_prefetch_b8).
        unsigned nrow = row + rows_per_iter;
        if (nrow < KTOP) {
            unsigned nidx = (unsigned)(smem[nrow] & 0xFFFFFFFFu);
            __builtin_prefetch(batch + (size_t)nidx * NFEAT, 0, 0);
        }
        const float4* src = (const float4*)(batch  + (size_t)idx * NFEAT);
        float4*       dst = (float4*)      (obatch + (size_t)row * NFEAT);
        dst[lane] = src[lane];
    }
}

extern "C" void kernel_launch(void* const* d_in, const int* in_sizes, int n_in,
                              void* d_out, int out_size, void* d_ws, size_t ws_size,
                              hipStream_t stream) {
    (void)in_sizes; (void)n_in; (void)d_ws; (void)ws_size; (void)out_size;
    const float* in  = (const float*)d_in[0];
    float*       out = (float*)d_out;
    topk_sort_gather_kernel<<<dim3(BATCHES), dim3(NTHREADS), 0, stream>>>(in, out);
}